// MultiVariateGaussian_11665131176505
// MI455X (gfx1250) — compile-verified
//
#include <hip/hip_runtime.h>

typedef float v2f __attribute__((ext_vector_type(2)));
typedef float v8f __attribute__((ext_vector_type(8)));

#define BSZ   64      // batch
#define CDIM  100     // channels
#define CPAD  112     // channels padded to 7*16 WMMA tiles
#define XS    66      // Xt row stride in floats (bank-conflict-free b64 loads)
#define CS    116     // cov row stride in floats (padded; 112 cols used)
#define EPSV  0.01f
#define NTHREADS 256

// ---------------------------------------------------------------------------
// Kernel 1: mean over batch. mean[c*P + p] = (1/64) sum_b emb[b][c][p].
// Streams the whole 80MB input once (coalesced) -> warms the 192MB L2 for k2.
// ---------------------------------------------------------------------------
__global__ __launch_bounds__(NTHREADS)
void mean_kernel(const float* __restrict__ emb, float* __restrict__ mean, int CP) {
    int idx = blockIdx.x * blockDim.x + threadIdx.x;
    if (idx >= CP) return;
    const float* q = emb + idx;
    float s = 0.f;
    #pragma unroll
    for (int b = 0; b < BSZ; ++b) s += q[(size_t)b * CP];
    mean[idx] = s * (1.0f / BSZ);
}

// ---------------------------------------------------------------------------
// Kernel 2: one workgroup per patch p (3136 blocks, 8 waves).
//   Stage A: centered X transposed into LDS: Xt[c][b], zero-padded c>=100
//   Stage B: cov = Xt*Xt^T/63 + eps*I via V_WMMA_F32_16X16X4_F32.
//            Only 28 upper-triangle 16x16 tiles computed; off-diagonal tiles
//            mirrored by a second LDS store (uniform branch, EXEC stays full).
//            cov buffer is padded to 112x116 so tile stores are unconditional
//            (no per-element exec-mask branches).
//   Stage C: in-place Gauss-Jordan inverse of the 100x100 region in LDS
//   Stage D: coalesced writeout
// ---------------------------------------------------------------------------
__global__ __launch_bounds__(NTHREADS)
void cov_inv_kernel(const float* __restrict__ emb,
                    const float* __restrict__ mean,
                    float* __restrict__ inv_out, int P) {
    extern __shared__ float lds[];
    float* meanL = lds;                    // CPAD
    float* Xt    = meanL + CPAD;           // CPAD * XS
    float* cov   = Xt + CPAD * XS;         // CPAD * CS   (padded tile region)
    float* rowk  = cov + CPAD * CS;        // CPAD
    float* colk  = rowk + CPAD;            // CPAD

    const int tid = threadIdx.x;
    const int p   = blockIdx.x;

    // ---- stage mean for this patch ----
    for (int c = tid; c < CDIM; c += NTHREADS)
        meanL[c] = mean[(size_t)c * P + p];
    __syncthreads();

    // ---- Stage A: centered, transposed X into LDS (zero pad c in [100,112)) ----
    for (int idx = tid; idx < BSZ * CPAD; idx += NTHREADS) {
        int b = idx / CPAD;
        int c = idx - b * CPAD;
        float v = 0.f;
        if (c < CDIM)
            v = emb[((size_t)b * CDIM + c) * P + p] - meanL[c];
        Xt[c * XS + b] = v;
    }
    __syncthreads();

    // ---- Stage B: WMMA covariance (upper triangle of 7x7 tile grid) ----
    const int lane  = tid & 31;
    const int wave  = tid >> 5;
    const int n16   = lane & 15;
    const int khalf = (lane < 16) ? 0 : 2;   // A/B fragment K half per ISA layout
    const int mofs  = (lane < 16) ? 0 : 8;   // C/D row offset per ISA layout

    for (int t = wave; t < 28; t += 8) {     // 28 = upper-triangle tiles incl diag
        // unrank t -> (ti, tj), ti <= tj  (uniform per wave)
        int ti = 0, rem = t;
        while (rem >= 7 - ti) { rem -= 7 - ti; ++ti; }
        const int tj = ti + rem;
        const int i0 = ti * 16;
        const int j0 = tj * 16;

        v8f acc = {};
        #pragma unroll
        for (int k0 = 0; k0 < BSZ; k0 += 4) {
            // A[m][k] = Xt[i0+m][k0+k] ; lanes 0-15 hold K=0,1; lanes 16-31 K=2,3
            v2f a = *(const v2f*)&Xt[(i0 + n16) * XS + k0 + khalf];
            // B[k][n] = Xt[j0+n][k0+k] ; mirrored lane/K split
            v2f b = *(const v2f*)&Xt[(j0 + n16) * XS + k0 + khalf];
            acc = __builtin_amdgcn_wmma_f32_16x16x4_f32(
                /*neg_a=*/false, a, /*neg_b=*/false, b,
                /*c_mod=*/(short)0, acc, /*reuse_a=*/false, /*reuse_b=*/false);
        }
        // store tile unconditionally into padded cov; mirror if off-diagonal.
        // VGPR r -> M=r (lanes 0-15) / M=r+8 (lanes 16-31), N = lane&15.
        #pragma unroll
        for (int r = 0; r < 8; ++r) {
            const int mi = i0 + r + mofs;
            const int nj = j0 + n16;
            float v = acc[r] * (1.0f / (BSZ - 1));
            v = (mi == nj) ? (v + EPSV) : v;       // cndmask, no branch
            cov[mi * CS + nj] = v;
            if (i0 != j0)                          // wave-uniform branch
                cov[nj * CS + mi] = v;
        }
    }
    __syncthreads();

    // ---- Stage C: in-place Gauss-Jordan inversion (no pivoting; SPD + eps*I) ----
    for (int k = 0; k < CDIM; ++k) {
        const float pinv = 1.0f / cov[k * CS + k];
        // stage scaled pivot row and pivot column into scratch (cov untouched)
        for (int j = tid; j < CDIM; j += NTHREADS) {
            float rv = (j == k) ? 1.0f : cov[k * CS + j];
            rowk[j] = rv * pinv;
            colk[j] = cov[j * CS + k];
        }
        __syncthreads();
        // eliminate
        for (int idx = tid; idx < CDIM * CDIM; idx += NTHREADS) {
            const int i = idx / CDIM;
            const int j = idx - i * CDIM;
            if (i == k) {
                cov[i * CS + j] = rowk[j];
            } else {
                const float f    = colk[i];
                const float base = (j == k) ? 0.0f : cov[i * CS + j];
                cov[i * CS + j]  = base - f * rowk[j];
            }
        }
        __syncthreads();
    }

    // ---- Stage D: writeout ----
    float* dst = inv_out + (size_t)p * (CDIM * CDIM);
    for (int idx = tid; idx < CDIM * CDIM; idx += NTHREADS) {
        const int i = idx / CDIM;
        const int j = idx - i * CDIM;
        dst[idx] = cov[i * CS + j];
    }
}

// ---------------------------------------------------------------------------
extern "C" void kernel_launch(void* const* d_in, const int* in_sizes, int n_in,
                              void* d_out, int out_size, void* d_ws, size_t ws_size,
                              hipStream_t stream) {
    const float* emb = (const float*)d_in[0];
    float* out = (float*)d_out;

    const int P  = 56 * 56;         // 3136 patches
    const int CP = CDIM * P;        // 313600

    float* mean_out = out;          // first C*P floats of output tuple
    float* inv_out  = out + CP;     // then P*C*C inverses

    // Kernel 1: mean (also primes L2 with the full 80MB input)
    mean_kernel<<<(CP + NTHREADS - 1) / NTHREADS, NTHREADS, 0, stream>>>(emb, mean_out, CP);

    // Kernel 2: fused covariance (WMMA, symmetric tiles) + inversion per patch
    const size_t ldsBytes =
        (size_t)(CPAD + CPAD * XS + CPAD * CS + 2 * CPAD) * sizeof(float); // ~83 KB
    cov_inv_kernel<<<P, NTHREADS, ldsBytes, stream>>>(emb, mean_out, inv_out, P);
}